// NerfRenderer_45638322487687
// MI455X (gfx1250) — compile-verified
//
#include <hip/hip_runtime.h>

// NeRF efficient-sampling + render, one ray per thread, wave32-per-block.
// Compute-bound on transcendental VALU; outputs (68 MB) live entirely in L2.
// gfx1250 paths used: global_load_async_to_lds_b32 (+ s_wait_asynccnt), wave32,
// large-LDS per-ray CDF scratch.

#define RPB      32          // rays (threads) per block = one wave32
#define NCOARSE  256
#define NFINE    128

#define F_LOG2E   1.4426950408889634f
#define F_LOG2_10 3.3219280948873623f
#define F_INV128  0.0078125f

#if defined(__has_builtin)
#  if __has_builtin(__builtin_amdgcn_global_load_async_to_lds_b32)
#    define HAVE_ASYNC_LDS 1
#  endif
#  if __has_builtin(__builtin_amdgcn_s_wait_asynccnt)
#    define HAVE_WAIT_ASYNC 1
#  endif
#endif

__device__ __forceinline__ float fexp2(float x) { return __builtin_amdgcn_exp2f(x); }
__device__ __forceinline__ float frcp (float x) { return __builtin_amdgcn_rcpf(x); }
__device__ __forceinline__ float fsq  (float x) { return __builtin_amdgcn_sqrtf(x); }
__device__ __forceinline__ float frsq (float x) { return __builtin_amdgcn_rsqf(x); }
__device__ __forceinline__ float sigm (float x) { return frcp(1.0f + fexp2(-x * F_LOG2E)); }

__global__ __launch_bounds__(RPB, 1)
void nerf_render_kernel(const float* __restrict__ rays_o,
                        const float* __restrict__ rays_d,
                        const float* __restrict__ bg,
                        float* __restrict__ out_image,
                        float* __restrict__ out_wf,
                        float* __restrict__ out_zvl,
                        float* __restrict__ out_inv,
                        int N)
{
    // Column layout [sample][ray]: lanes hit consecutive DWORDs -> 0 bank conflicts.
    __shared__ float s_cdf[NCOARSE * RPB];
    __shared__ float s_ray[6 * RPB];

    const int tid = threadIdx.x;
    const int ray = blockIdx.x * RPB + tid;
    if (ray >= N) return;

    const float* go = rays_o + (size_t)ray * 3;
    const float* gd = rays_d + (size_t)ray * 3;

    float ox, oy, oz, dx, dy, dz;
#if defined(HAVE_ASYNC_LDS)
    {   // gfx1250 async global->LDS staging (tracked by ASYNCcnt)
        typedef __attribute__((address_space(1))) int gint_t;   // global src
        typedef __attribute__((address_space(3))) int lint_t;   // LDS dst
        gint_t* g0 = (gint_t*)go;  lint_t* l0 = (lint_t*)&s_ray[tid * 6];
        gint_t* g1 = (gint_t*)gd;  lint_t* l1 = (lint_t*)&s_ray[tid * 6 + 3];
        __builtin_amdgcn_global_load_async_to_lds_b32(g0, l0, 0, 0);
        __builtin_amdgcn_global_load_async_to_lds_b32(g0, l0, 4, 0);
        __builtin_amdgcn_global_load_async_to_lds_b32(g0, l0, 8, 0);
        __builtin_amdgcn_global_load_async_to_lds_b32(g1, l1, 0, 0);
        __builtin_amdgcn_global_load_async_to_lds_b32(g1, l1, 4, 0);
        __builtin_amdgcn_global_load_async_to_lds_b32(g1, l1, 8, 0);
    }
#  if defined(HAVE_WAIT_ASYNC)
    __builtin_amdgcn_s_wait_asynccnt(0);
    asm volatile("" ::: "memory");
#  else
    asm volatile("s_wait_asynccnt 0" ::: "memory");
#  endif
    ox = s_ray[tid*6+0]; oy = s_ray[tid*6+1]; oz = s_ray[tid*6+2];
    dx = s_ray[tid*6+3]; dy = s_ray[tid*6+4]; dz = s_ray[tid*6+5];
#else
    __builtin_prefetch(go, 0, 0);   // global_prefetch_b8
    __builtin_prefetch(gd, 0, 0);
    ox = go[0]; oy = go[1]; oz = go[2];
    dx = gd[0]; dy = gd[1]; dz = gd[2];
#endif

    // normalize direction
    {
        float rn = frsq(dx*dx + dy*dy + dz*dz);
        dx *= rn; dy *= rn; dz *= rn;
    }

    // ---------- pass 1: coarse density -> raw weights (sequential transmittance) ----
    // Only T carries a loop dependence; unrolling overlaps 4 independent trans chains.
    {
        float T = 1.0f;
#pragma unroll 4
        for (int i = 0; i < NCOARSE; ++i) {
            // zl_i = (i-128)/128 exactly (uniform log grid); zf = 10^zl
            float zf = fexp2((float)(i - 128) * (F_INV128 * F_LOG2_10));
            float px = fmaf(dx, zf, ox), py = fmaf(dy, zf, oy), pz = fmaf(dz, zf, oz);
            float rr = px*px + py*py + pz*pz;
            float nr = fsq(rr);
            float in_ = frcp(fmaxf(nr, 1e-8f));
            float kk  = (nr <= 1.0f) ? 1.0f : (2.0f - in_) * in_;   // mip360 contraction
            float sx = px*kk, sy = py*kk, sz = pz*kk;
            float s2 = sx*sx + sy*sy + sz*sz;
            float sig = 25.0f * fexp2(-4.0f * F_LOG2E * s2);
            float delta = (i < NCOARSE - 1) ? F_INV128 : 0.0f;      // last delta = 0
            float e = fexp2(-sig * delta * F_LOG2E);                // 1 - alpha
            s_cdf[i * RPB + tid] = (1.0f - e) * T;                  // weight
            T *= e;
        }
    }

    // ---------- pass 2: max-filter reweight + bias, unnormalized CDF in place ------
    float csum;
    {
        float wprev = 0.0f;
        float wcur  = s_cdf[tid];
        float c = 0.0f;
        const float bias = 1e-4f / 256.0f;   // ALPHA_RW / S (scale=0.5 cancels)
#pragma unroll 4
        for (int i = 0; i < NCOARSE; ++i) {
            float wnext = (i < NCOARSE - 1) ? s_cdf[(i + 1) * RPB + tid] : 0.0f;
            float wr = 0.5f * (fmaxf(wprev, wcur) + fmaxf(wcur, wnext)) + bias;
            c += wr;
            s_cdf[i * RPB + tid] = c;        // overwrite w[i] after w[i+1] was read
            wprev = wcur; wcur = wnext;
        }
        csum = c;
    }

    // ---------- pass 3: monotone inverse-CDF merge fused with fine render ----------
    const float bg0 = bg[0], bg1 = bg[1], bg2 = bg[2];
    float T = 1.0f, im0 = 0.0f, im1 = 0.0f, im2 = 0.0f, invd = 0.0f, wsum = 0.0f;
    int ind = 0;                               // searchsorted-right pointer (monotone u)
    const float eps_thr = 1e-8f * csum;        // normalized 1e-8 guard, unnormalized form

    auto sample_z = [&](float u) -> float {
        float ut = u * csum;                   // compare against unnormalized cdf
        while (ind < NCOARSE && s_cdf[ind * RPB + tid] <= ut) ++ind;
        int below = (ind > 0) ? ind - 1 : 0;
        int above = (ind < NCOARSE) ? ind : NCOARSE - 1;
        float c0 = s_cdf[below * RPB + tid];
        float c1 = s_cdf[above * RPB + tid];
        float dc = c1 - c0;
        float denom = (dc < eps_thr) ? csum : dc;  // matches denom=1 in normalized space
        float t = fminf(fmaxf((ut - c0) * frcp(denom), 0.0f), 1.0f);
        return ((float)(below - 128) + t * (float)(above - below)) * F_INV128;
    };

    float* wf_out  = out_wf  + (size_t)ray * NFINE;
    float* zvl_out = out_zvl + (size_t)ray * NFINE;

    float zlf = sample_z(0.5f / (float)NFINE);
    for (int k = 0; k < NFINE; ++k) {
        float zlf_next = (k < NFINE - 1) ? sample_z(((float)k + 1.5f) / (float)NFINE) : zlf;
        float delta = zlf_next - zlf;                      // 0 on last sample
        float zf = fexp2(zlf * F_LOG2_10);
        float px = fmaf(dx, zf, ox), py = fmaf(dy, zf, oy), pz = fmaf(dz, zf, oz);
        float rr = px*px + py*py + pz*pz;
        float nr = fsq(rr);
        float in_ = frcp(fmaxf(nr, 1e-8f));
        float kk  = (nr <= 1.0f) ? 1.0f : (2.0f - in_) * in_;
        float sx = px*kk, sy = py*kk, sz = pz*kk;
        float s2 = sx*sx + sy*sy + sz*sz;
        float sig = 25.0f * fexp2(-4.0f * F_LOG2E * s2);
        float c0 = sigm(fmaf(0.1f, dx, sx));
        float c1 = sigm(fmaf(0.1f, dy, sy));
        float c2 = sigm(fmaf(0.1f, dz, sz));
        float e = fexp2(-sig * delta * F_LOG2E);
        float w = (1.0f - e) * T;
        T *= e;
        im0 = fmaf(w, c0, im0);
        im1 = fmaf(w, c1, im1);
        im2 = fmaf(w, c2, im2);
        invd = fmaf(w, frcp(zf), invd);
        wsum += w;
        wf_out[k]  = w;
        zvl_out[k] = fmaf(zlf, 0.5f, 0.5f);   // (zlf - lo)/(hi - lo), lo=-1, hi=1
        zlf = zlf_next;
    }

    out_image[(size_t)ray * 3 + 0] = fmaf(1.0f - wsum, bg0, im0);
    out_image[(size_t)ray * 3 + 1] = fmaf(1.0f - wsum, bg1, im1);
    out_image[(size_t)ray * 3 + 2] = fmaf(1.0f - wsum, bg2, im2);
    out_inv[ray] = invd;
}

extern "C" void kernel_launch(void* const* d_in, const int* in_sizes, int n_in,
                              void* d_out, int out_size, void* d_ws, size_t ws_size,
                              hipStream_t stream)
{
    (void)n_in; (void)out_size; (void)d_ws; (void)ws_size;
    const float* rays_o = (const float*)d_in[0];
    const float* rays_d = (const float*)d_in[1];
    const float* bg     = (const float*)d_in[2];
    const int N = in_sizes[0] / 3;

    float* out       = (float*)d_out;
    float* out_image = out;                       // [N,3]
    float* out_wf    = out + (size_t)3   * N;     // [N,128]
    float* out_zvl   = out + (size_t)131 * N;     // [N,128]
    float* out_inv   = out + (size_t)259 * N;     // [N]

    dim3 grid((N + RPB - 1) / RPB), block(RPB);
    hipLaunchKernelGGL(nerf_render_kernel, grid, block, 0, stream,
                       rays_o, rays_d, bg, out_image, out_wf, out_zvl, out_inv, N);
}